// Model_39676907887639
// MI455X (gfx1250) — compile-verified
//
#include <hip/hip_runtime.h>
#include <stdint.h>

// ---------------------------------------------------------------------------
// MI455X (gfx1250) implementation.
//
// The reference DAG reduces (after dead-code elimination and cat/crop algebra)
// to a chain of shared-weight 3x3 depthwise convs + scalar-weighted adds on
// 192x192 f32 planes, finishing with a 24-channel scaled broadcast of
// g(m) = dwconv(m) + m over 4 base maps per batch.
//
// Bandwidth-bound (~0.33 GFLOP vs >=150 MB traffic), so the design is:
//  * all intermediates in a 36 MB workspace -> L2-resident (192 MB L2)
//  * async global->LDS staging of conv tiles (ASYNCcnt path, CDNA5)
//  * conv evaluated per 16x16 tile as banded matmuls on V_WMMA_F32_16X16X4_F32
//    (15 WMMAs/tile: 3 weight rows x 5 K-chunks of a 16x20 * 20x16 product)
//  * non-temporal stores for the 113 MB write-once output
// ---------------------------------------------------------------------------

typedef __attribute__((ext_vector_type(2))) float v2f;
typedef __attribute__((ext_vector_type(8))) float v8f;

#define IMG_H 192
#define IMG_W 192
#define PLANE (IMG_H * IMG_W) // 36864 floats per plane

// Per-block output region: 32 rows x 64 cols (8 waves, one 16x16 WMMA tile each,
// arranged 2x4). Input staged with 1-px halo: 34 x 66, pitch 68 (bank-friendly;
// pad columns 66..67 stay zero and double as the K=18..19 zero-pad of the matmul).
#define LROWS 34
#define LCOLS 66
#define LPITCH 68
#define LELEMS (LROWS * LPITCH) // 2312 floats = 9248 B

// ---------- async global -> LDS staging (CDNA5 ASYNCcnt path) --------------
// Zero the tile (SAME-conv zero padding + pitch pad), then gather the valid
// region with per-lane global_load_async_to_lds_b32. LDS byte offset = low 32
// bits of the generic pointer (flat LDS aperture puts the wave-relative LDS
// address in addr[31:0]).
__device__ __forceinline__ void stage_tile(const float* __restrict__ plane,
                                           float* lds, int h0, int w0)
{
  const int tid = threadIdx.x;
  for (int i = tid; i < LELEMS; i += 256) lds[i] = 0.0f;
  __syncthreads();
  for (int i = tid; i < LROWS * LCOLS; i += 256) {
    const int rr = i / LCOLS;
    const int cc = i - rr * LCOLS;
    const int gr = h0 - 1 + rr;
    const int gc = w0 - 1 + cc;
    if ((unsigned)gr < (unsigned)IMG_H && (unsigned)gc < (unsigned)IMG_W) {
      unsigned ldsb = (unsigned)(uintptr_t)(lds + rr * LPITCH + cc);
      const float* gp = plane + gr * IMG_W + gc;
      asm volatile("global_load_async_to_lds_b32 %0, %1, off"
                   :: "v"(ldsb), "v"(gp)
                   : "memory");
    }
  }
  asm volatile("s_wait_asynccnt 0x0" ::: "memory");
  __syncthreads();
}

// ---------- banded-Toeplitz B fragments for the conv-as-matmul -------------
// Out[M,N] = sum_k A[M,k]*B[k,N], A[M,k] = in[r0+dh-1+M, c0-1+k] (k in 0..17,
// 18..19 zero-padded), B[k,N] = W[dh, k-N] for k-N in {0,1,2} else 0.
// f32 operand striping (per ISA 16x4 A layout; B assumed symmetric):
// lane half (lane>=16) holds K = 4*chunk + 2*half + {0,1} in V0/V1.
__device__ __forceinline__ void make_bfrags(const float* w9, v2f* bf)
{
  const int lane = threadIdx.x & 31;
  const int half = lane >> 4;
  const int n = lane & 15;
  int idx = 0;
  #pragma unroll
  for (int dh = 0; dh < 3; ++dh) {
    const float r0w = w9[dh * 3 + 0];
    const float r1w = w9[dh * 3 + 1];
    const float r2w = w9[dh * 3 + 2];
    #pragma unroll
    for (int ch = 0; ch < 5; ++ch) {
      const int kx = 4 * ch + 2 * half;
      const int ky = kx + 1;
      const int dx = kx - n;
      const int dy = ky - n;
      v2f b;
      b.x = (dx == 0) ? r0w : (dx == 1) ? r1w : (dx == 2) ? r2w : 0.0f;
      b.y = (dy == 0) ? r0w : (dy == 1) ? r1w : (dy == 2) ? r2w : 0.0f;
      bf[idx++] = b;
    }
  }
}

// ---------- one 16x16 conv tile via 15x V_WMMA_F32_16X16X4_F32 -------------
// Output element (M,N) of the tile at LDS coords (r0,c0); needs LDS rows
// r0-1..r0+16, cols c0-1..c0+18 (all in-bounds & zero-padded by stage_tile).
__device__ __forceinline__ v8f conv16x16(const float* lds, int r0, int c0,
                                         const v2f* bf)
{
  const int lane = threadIdx.x & 31;
  const int half = lane >> 4;
  const int m = lane & 15;
  v8f acc = {0.f, 0.f, 0.f, 0.f, 0.f, 0.f, 0.f, 0.f};
  int idx = 0;
  #pragma unroll
  for (int dh = 0; dh < 3; ++dh) {
    const float* rowp = lds + (r0 + dh - 1 + m) * LPITCH + (c0 - 1);
    #pragma unroll
    for (int ch = 0; ch < 5; ++ch) {
      const int k0 = 4 * ch + 2 * half;
      v2f a;
      a.x = rowp[k0];
      a.y = rowp[k0 + 1];
      acc = __builtin_amdgcn_wmma_f32_16x16x4_f32(
          false, a, false, bf[idx], (short)0, acc, false, false);
      ++idx;
    }
  }
  return acc;
}

// ---------- generic plane op: out = cc*(dwconv(src)+bias) + cs*src + co*oth
// grid = (3, 6, nPlanes); block = 256 (8 waves, one 16x16 tile each).
// outStride is in planes (lets us scatter into the interleaved w[32,2] buffer).
__global__ __launch_bounds__(256) void k_conv(
    const float* __restrict__ src, const float* __restrict__ oth,
    const float* __restrict__ Wg, const float* __restrict__ Bg,
    float* __restrict__ out1, int out1Stride, float c1c, float c1s, float c1o,
    float* __restrict__ out2, int out2Stride, float c2c, float c2s, float c2o)
{
  __shared__ float lds[LELEMS];
  const int n = blockIdx.z;
  const int h0 = 32 * blockIdx.y;
  const int w0 = 64 * blockIdx.x;

  float w9[9];
  #pragma unroll
  for (int i = 0; i < 9; ++i) w9[i] = Wg[i];
  const float bias = Bg[0];

  v2f bf[15];
  make_bfrags(w9, bf);
  stage_tile(src + (size_t)n * PLANE, lds, h0, w0);

  const int tid = threadIdx.x;
  const int wave = tid >> 5, lane = tid & 31;
  const int half = lane >> 4, ncol = lane & 15;
  const int tr = wave >> 2, tc = wave & 3;
  const int r0 = 1 + 16 * tr, c0 = 1 + 16 * tc;

  const v8f acc = conv16x16(lds, r0, c0, bf);

  #pragma unroll
  for (int r = 0; r < 8; ++r) {
    const int M = r + 8 * half; // C/D layout: reg r, lanes>=16 -> row r+8
    const int gh = h0 + 16 * tr + M;
    const int gw = w0 + 16 * tc + ncol;
    const size_t off = (size_t)gh * IMG_W + gw;
    const float center = lds[(r0 + M) * LPITCH + (c0 + ncol)];
    const float cv = acc[r] + bias;
    float ov = 0.0f;
    if (oth) ov = oth[(size_t)n * PLANE + off];
    out1[(size_t)n * out1Stride * PLANE + off] = c1c * cv + c1s * center + c1o * ov;
    if (out2)
      out2[(size_t)n * out2Stride * PLANE + off] = c2c * cv + c2s * center + c2o * ov;
  }
}

// ---------- finale: g = dwconv(m)+m, fan out 24 scaled channels + bias -----
// grid = (3, 6, 32*4); z -> (n = z>>2, map c = z&3); maps {p4_0,p4_1,q_0,q_1}.
// Channel 6*blk+c gets alpha2[blk]*g + b; q maps also feed 6*blk+c+2 with
// alpha1[blk]*g + b  (alpha2 = 2*alpha, alpha = [6,3,6,3]).
__global__ __launch_bounds__(256) void k_final(
    const float* __restrict__ p4, const float* __restrict__ qB,
    const float* __restrict__ Wg, const float* __restrict__ Bg,
    float* __restrict__ out)
{
  __shared__ float lds[LELEMS];
  const int z = blockIdx.z;
  const int n = z >> 2;
  const int c = z & 3;
  const int h0 = 32 * blockIdx.y;
  const int w0 = 64 * blockIdx.x;
  const float* mp = (c < 2) ? (p4 + (size_t)(2 * n + c) * PLANE)
                            : (qB + (size_t)(2 * n + (c - 2)) * PLANE);

  float w9[9];
  #pragma unroll
  for (int i = 0; i < 9; ++i) w9[i] = Wg[i];
  const float bias = Bg[0];

  v2f bf[15];
  make_bfrags(w9, bf);
  stage_tile(mp, lds, h0, w0);

  const int tid = threadIdx.x;
  const int wave = tid >> 5, lane = tid & 31;
  const int half = lane >> 4, ncol = lane & 15;
  const int tr = wave >> 2, tc = wave & 3;
  const int r0 = 1 + 16 * tr, c0 = 1 + 16 * tc;

  const v8f acc = conv16x16(lds, r0, c0, bf);

  const float a2[4] = {12.f, 6.f, 12.f, 6.f};
  const float a1[4] = {6.f, 3.f, 6.f, 3.f};

  #pragma unroll
  for (int r = 0; r < 8; ++r) {
    const int M = r + 8 * half;
    const int gh = h0 + 16 * tr + M;
    const int gw = w0 + 16 * tc + ncol;
    const size_t off = (size_t)gh * IMG_W + gw;
    const float g = acc[r] + lds[(r0 + M) * LPITCH + (c0 + ncol)];
    #pragma unroll
    for (int blk = 0; blk < 4; ++blk) {
      __builtin_nontemporal_store(
          a2[blk] * g + bias, out + (size_t)(24 * n + 6 * blk + c) * PLANE + off);
      if (c >= 2)
        __builtin_nontemporal_store(
            a1[blk] * g + bias,
            out + (size_t)(24 * n + 6 * blk + c + 2) * PLANE + off);
    }
  }
}

// ---------------------------------------------------------------------------
extern "C" void kernel_launch(void* const* d_in, const int* in_sizes, int n_in,
                              void* d_out, int out_size, void* d_ws, size_t ws_size,
                              hipStream_t stream)
{
  (void)in_sizes; (void)n_in; (void)out_size; (void)ws_size;

  const float* x1 = (const float*)d_in[0];
  const float* x2 = (const float*)d_in[1];
  const float* x3 = (const float*)d_in[2];
  const float* Wg = (const float*)d_in[3]; // 9 floats
  const float* Bg = (const float*)d_in[4]; // 1 float
  float* out = (float*)d_out;
  float* ws = (float*)d_ws;

  // Workspace: 256 planes = 36 MB (fits comfortably in 192 MB L2).
  //  region0 (64P): e1|e2|s|v0  -> later h -> later p4
  //  region1 (64P): w           -> later t
  //  pB (64P), qB (64P)
  float* r0_ = ws;
  float* r1_ = ws + (size_t)64 * PLANE;
  float* pB  = ws + (size_t)128 * PLANE;
  float* qB  = ws + (size_t)192 * PLANE;

  float* e1 = r0_;
  float* e2 = r0_ + (size_t)16 * PLANE;
  float* sB = r0_ + (size_t)32 * PLANE;
  float* v0 = r0_ + (size_t)48 * PLANE;
  float* wB = r1_;
  float* hB = r0_; // e1/e2/s/v0 dead by then
  float* tB = r1_; // w dead by then
  float* p4 = r0_; // h dead by then

  const dim3 blk(256), g16(3, 6, 16), g64(3, 6, 64), gF(3, 6, 128);

  // 1) e1 = conv(x1) + x2
  k_conv<<<g16, blk, 0, stream>>>(x1, x2, Wg, Bg, e1, 1, 1.f, 0.f, 1.f,
                                  nullptr, 0, 0.f, 0.f, 0.f);
  // 2) e2 = conv(x2) + x3
  k_conv<<<g16, blk, 0, stream>>>(x2, x3, Wg, Bg, e2, 1, 1.f, 0.f, 1.f,
                                  nullptr, 0, 0.f, 0.f, 0.f);
  // 3) w[n,1] = conv(x3)          (planes 2n+1)
  k_conv<<<g16, blk, 0, stream>>>(x3, nullptr, Wg, Bg, wB + PLANE, 2, 1.f, 0.f, 0.f,
                                  nullptr, 0, 0.f, 0.f, 0.f);
  // 4) s = conv(e1)
  k_conv<<<g16, blk, 0, stream>>>(e1, nullptr, Wg, Bg, sB, 1, 1.f, 0.f, 0.f,
                                  nullptr, 0, 0.f, 0.f, 0.f);
  // 5) w[n,0] = conv(s); v0 = s + conv(s)
  k_conv<<<g16, blk, 0, stream>>>(sB, nullptr, Wg, Bg, wB, 2, 1.f, 0.f, 0.f,
                                  v0, 1, 1.f, 1.f, 0.f);
  // 6) w[16+n,0] = conv(v0)       (planes 32+2n)
  k_conv<<<g16, blk, 0, stream>>>(v0, nullptr, Wg, Bg,
                                  wB + (size_t)32 * PLANE, 2, 1.f, 0.f, 0.f,
                                  nullptr, 0, 0.f, 0.f, 0.f);
  // 7) w[16+n,1] = conv(e2)       (planes 33+2n)
  k_conv<<<g16, blk, 0, stream>>>(e2, nullptr, Wg, Bg,
                                  wB + (size_t)33 * PLANE, 2, 1.f, 0.f, 0.f,
                                  nullptr, 0, 0.f, 0.f, 0.f);
  // 8) p = w + conv(w);  q = w + 3*conv(w)
  k_conv<<<g64, blk, 0, stream>>>(wB, nullptr, Wg, Bg, pB, 1, 1.f, 1.f, 0.f,
                                  qB, 1, 3.f, 1.f, 0.f);
  // 9) h = conv(q) + p
  k_conv<<<g64, blk, 0, stream>>>(qB, pB, Wg, Bg, hB, 1, 1.f, 0.f, 1.f,
                                  nullptr, 0, 0.f, 0.f, 0.f);
  // 10) t = conv(h) + p
  k_conv<<<g64, blk, 0, stream>>>(hB, pB, Wg, Bg, tB, 1, 1.f, 0.f, 1.f,
                                  nullptr, 0, 0.f, 0.f, 0.f);
  // 11) p4 = conv(t) + 2*t + p
  k_conv<<<g64, blk, 0, stream>>>(tB, pB, Wg, Bg, p4, 1, 1.f, 2.f, 1.f,
                                  nullptr, 0, 0.f, 0.f, 0.f);
  // 12) out[n, 6*blk+c] = scale * (dwconv(m_c) + m_c) + b
  k_final<<<gF, blk, 0, stream>>>(p4, qB, Wg, Bg, out);
}